// TorchReRaLSTM_35536559407387
// MI455X (gfx1250) — compile-verified
//
#include <hip/hip_runtime.h>
#include <hip/hip_bf16.h>

#define N 2048
#define U 64
#define R 64

typedef float v2f __attribute__((ext_vector_type(2)));
typedef float v4f __attribute__((ext_vector_type(4)));
typedef float v8f __attribute__((ext_vector_type(8)));

// ---------------------------------------------------------------------------
// K1: head[i] = relu(feature[i,:]@head_w + head_b); tail likewise.
// ---------------------------------------------------------------------------
__global__ void k_headtail(const float* __restrict__ feature,
                           const float* __restrict__ head_w, const float* __restrict__ head_b,
                           const float* __restrict__ tail_w, const float* __restrict__ tail_b,
                           float* __restrict__ head, float* __restrict__ tail) {
    int i = blockIdx.x * blockDim.x + threadIdx.x;
    if (i >= N) return;
    const float* f = feature + (size_t)i * U;
    float h = 0.f, t = 0.f;
    for (int u = 0; u < U; ++u) {
        float x = f[u];
        h += x * head_w[u];
        t += x * tail_w[u];
    }
    head[i] = fmaxf(h + head_b[0], 0.0f);
    tail[i] = fmaxf(t + tail_b[0], 0.0f);
}

// ---------------------------------------------------------------------------
// K2: S[i,j] = rel_mask[i,j] + head[i] + tail[j] + relu(relation[i,j,:]@rel_w + rel_b)
// The 1 GiB stream. Fully coalesced: wave loads 32 x float4 = 512B = two (i,j)
// pairs per iteration; 16-lane xor-shuffle reduction for each 64-wide dot.
// relation/rel_mask read exactly once -> non-temporal (keep L2 for S).
// ---------------------------------------------------------------------------
__global__ void k_scores(const v4f* __restrict__ rel4,
                         const float* __restrict__ rel_w,
                         const float* __restrict__ rel_b,
                         const float* __restrict__ rel_mask,
                         const float* __restrict__ head,
                         const float* __restrict__ tail,
                         float* __restrict__ S) {
    const int lane = threadIdx.x & 31;
    const int c    = lane & 15;                     // 16 lanes cover one 64-float row
    const v4f rw   = *(const v4f*)(rel_w + c * 4);  // each lane's rel_w chunk
    const float rb = rel_b[0];

    const long nwaves = (long)((gridDim.x * blockDim.x) >> 5);
    const long wave   = ((long)blockIdx.x * blockDim.x + threadIdx.x) >> 5;
    const long NP     = (long)N * N;

    for (long p0 = wave * 2; p0 < NP; p0 += nwaves * 2) {
        long p = p0 + (lane >> 4);                  // pair handled by this half-wave
        v4f r = __builtin_nontemporal_load(rel4 + p * 16 + c);
        float s = r.x * rw.x + r.y * rw.y + r.z * rw.z + r.w * rw.w;
        s += __shfl_xor(s, 1, 32);
        s += __shfl_xor(s, 2, 32);
        s += __shfl_xor(s, 4, 32);
        s += __shfl_xor(s, 8, 32);                  // lanes 0/16 now hold full dots
        if (c == 0) {
            int i = (int)(p >> 11);                 // N = 2048 = 2^11
            int j = (int)(p & (N - 1));
            float rm = __builtin_nontemporal_load(rel_mask + p);
            S[p] = rm + head[i] + tail[j] + fmaxf(s + rb, 0.0f);
        }
    }
}

// ---------------------------------------------------------------------------
// K3a: per-(row-segment, column) online softmax partials over axis 0.
// grid = 8 row segments x 8 column groups; coalesced column-major reads.
// ---------------------------------------------------------------------------
__global__ void k_colstats_partial(const float* __restrict__ S,
                                   float* __restrict__ pm, float* __restrict__ ps) {
    int j  = (blockIdx.x & 7) * 256 + threadIdx.x;  // column
    int rs = blockIdx.x >> 3;                       // row segment 0..7
    float m = -3.0e38f, sm = 0.0f;
    int rend = rs * 256 + 256;
    for (int r = rs * 256; r < rend; ++r) {
        float x  = S[(size_t)r * N + j];
        float nm = fmaxf(m, x);
        sm = sm * __expf(m - nm) + __expf(x - nm);
        m  = nm;
    }
    pm[rs * N + j] = m;
    ps[rs * N + j] = sm;
}

// K3b: combine 8 partials -> colmax[j], cinv[j] = 1/Z[j]
__global__ void k_colstats_final(const float* __restrict__ pm, const float* __restrict__ ps,
                                 float* __restrict__ colmax, float* __restrict__ cinv) {
    int j = blockIdx.x * blockDim.x + threadIdx.x;
    if (j >= N) return;
    float m = -3.0e38f;
    for (int rs = 0; rs < 8; ++rs) m = fmaxf(m, pm[rs * N + j]);
    float Z = 0.0f;
    for (int rs = 0; rs < 8; ++rs) Z += ps[rs * N + j] * __expf(pm[rs * N + j] - m);
    colmax[j] = m;
    cinv[j]   = 1.0f / Z;
}

// ---------------------------------------------------------------------------
// K4: outputs_proped = softmax(S, axis=0) @ feature  via V_WMMA_F32_16X16X4_F32.
// Block = 4 waves; wave w owns output tile [i0:i0+16, 16w:16w+16], K = 2048 in
// steps of 4. Softmax normalization applied on the fly in the A-load.
// A 16x4 layout (ISA 7.12.2): lanes 0-15 -> {K=0,K=1}, lanes 16-31 -> {K=2,K=3}.
// B 4x16 mirrored; D: VGPR v -> M=v (lanes 0-15) / M=8+v (lanes 16-31).
// ---------------------------------------------------------------------------
__global__ void k_aggregate_wmma(const float* __restrict__ S,
                                 const float* __restrict__ colmax,
                                 const float* __restrict__ cinv,
                                 const float* __restrict__ feature,
                                 float* __restrict__ outp) {
    const int lane = threadIdx.x & 31;
    const int half = lane >> 4;
    const int l15  = lane & 15;
    const int u0   = (threadIdx.x >> 5) * 16;
    const int i0   = blockIdx.x * 16;
    const int row  = i0 + l15;
    const int koff = half * 2;

    v8f acc = {};
    for (int j = 0; j < N; j += 4) {
        int k0 = j + koff;
        v2f sp = *(const v2f*)(S + (size_t)row * N + k0);
        v2f cm = *(const v2f*)(colmax + k0);
        v2f ci = *(const v2f*)(cinv + k0);
        v2f a;
        a.x = __expf(sp.x - cm.x) * ci.x;
        a.y = __expf(sp.y - cm.y) * ci.y;
        v2f b;
        b.x = feature[(size_t)(k0)     * U + u0 + l15];
        b.y = feature[(size_t)(k0 + 1) * U + u0 + l15];
        acc = __builtin_amdgcn_wmma_f32_16x16x4_f32(false, a, false, b,
                                                    (short)0, acc, false, false);
    }
    for (int v = 0; v < 8; ++v) {
        int r = i0 + v + half * 8;
        outp[(size_t)r * U + u0 + l15] = acc[v];
    }
}

// ---------------------------------------------------------------------------
// K5: prediction = leaky_relu([feature, outp] @ pred_w + pred_b);
//     rr = (prediction - base_price) / base_price
// ---------------------------------------------------------------------------
__global__ void k_predict(const float* __restrict__ feature, const float* __restrict__ outp,
                          const float* __restrict__ pred_w, const float* __restrict__ pred_b,
                          const float* __restrict__ base_price, float* __restrict__ rr) {
    int i = blockIdx.x * blockDim.x + threadIdx.x;
    if (i >= N) return;
    const float* f = feature + (size_t)i * U;
    const float* o = outp    + (size_t)i * U;
    float s = pred_b[0];
    for (int u = 0; u < U; ++u) s += f[u] * pred_w[u];
    for (int u = 0; u < U; ++u) s += o[u] * pred_w[U + u];
    float pred = (s >= 0.0f) ? s : 0.2f * s;
    float bp = base_price[i];
    rr[i] = (pred - bp) / bp;
}

// ---------------------------------------------------------------------------
// K6: reg_loss = sum((rr-gt)^2);
//     rank_loss = mean_{ij} relu(rr_i*rr_j*m_i*m_j) = (P^2 + Q^2)/N^2
//     with P/Q = sums of positive/negative rr*m (exact O(N) reformulation).
// ---------------------------------------------------------------------------
__global__ void k_loss(const float* __restrict__ rr, const float* __restrict__ gt,
                       const float* __restrict__ mask, float* __restrict__ out) {
    __shared__ float sS[256], sP[256], sQ[256];
    int t = threadIdx.x;
    float ssq = 0.f, P = 0.f, Q = 0.f;
    for (int i = t; i < N; i += 256) {
        float d = rr[i] - gt[i];
        ssq += d * d;
        float x = rr[i] * mask[i];
        if (x > 0.0f) P += x; else Q += x;
    }
    sS[t] = ssq; sP[t] = P; sQ[t] = Q;
    __syncthreads();
    for (int off = 128; off > 0; off >>= 1) {
        if (t < off) { sS[t] += sS[t+off]; sP[t] += sP[t+off]; sQ[t] += sQ[t+off]; }
        __syncthreads();
    }
    if (t == 0) {
        float reg  = sS[0];
        float rank = (sP[0] * sP[0] + sQ[0] * sQ[0]) / ((float)N * (float)N);
        out[0] = reg + 1.0f * rank;   // ALPHA = 1.0
        out[1] = reg;
        out[2] = rank;
    }
}

// ---------------------------------------------------------------------------
extern "C" void kernel_launch(void* const* d_in, const int* in_sizes, int n_in,
                              void* d_out, int out_size, void* d_ws, size_t ws_size,
                              hipStream_t stream) {
    const float* feature    = (const float*)d_in[0];
    const float* relation   = (const float*)d_in[1];
    const float* rel_mask   = (const float*)d_in[2];
    const float* base_price = (const float*)d_in[3];
    const float* ground_tr  = (const float*)d_in[4];
    const float* mask       = (const float*)d_in[5];
    const float* rel_w      = (const float*)d_in[6];
    const float* rel_b      = (const float*)d_in[7];
    const float* head_w     = (const float*)d_in[8];
    const float* head_b     = (const float*)d_in[9];
    const float* tail_w     = (const float*)d_in[10];
    const float* tail_b     = (const float*)d_in[11];
    const float* pred_w     = (const float*)d_in[12];
    const float* pred_b     = (const float*)d_in[13];
    float* out = (float*)d_out;

    // workspace layout (floats); total ~17.5 MB
    float* ws     = (float*)d_ws;
    float* S      = ws;                            // N*N      = 4,194,304
    float* head   = S + (size_t)N * N;             // 2048
    float* tail   = head + N;                      // 2048
    float* colmax = tail + N;                      // 2048
    float* cinv   = colmax + N;                    // 2048
    float* pm     = cinv + N;                      // 8*N = 16384
    float* ps     = pm + 8 * N;                    // 16384
    float* outp   = ps + 8 * N;                    // N*U = 131,072
    float* rr     = outp + (size_t)N * U;          // 2048

    k_headtail<<<N / 256, 256, 0, stream>>>(feature, head_w, head_b, tail_w, tail_b, head, tail);
    k_scores<<<2048, 256, 0, stream>>>((const v4f*)relation, rel_w, rel_b, rel_mask, head, tail, S);
    k_colstats_partial<<<64, 256, 0, stream>>>(S, pm, ps);
    k_colstats_final<<<N / 256, 256, 0, stream>>>(pm, ps, colmax, cinv);
    k_aggregate_wmma<<<N / 16, 128, 0, stream>>>(S, colmax, cinv, feature, outp);
    k_predict<<<N / 256, 256, 0, stream>>>(feature, outp, pred_w, pred_b, base_price, rr);
    k_loss<<<1, 256, 0, stream>>>(rr, ground_tr, mask, out);
}